// BaseTopologicalLayer_44727789421014
// MI455X (gfx1250) — compile-verified
//
#include <hip/hip_runtime.h>
#include <stdint.h>

#define NPTS 4096
#define THREADS 1024
#define PER 4  // NPTS / THREADS

typedef unsigned int u32x4 __attribute__((ext_vector_type(4)));
typedef int i32x4 __attribute__((ext_vector_type(4)));
typedef int i32x8 __attribute__((ext_vector_type(8)));

#if defined(__HIP_DEVICE_COMPILE__) && __has_builtin(__builtin_amdgcn_tensor_load_to_lds) && __has_builtin(__builtin_amdgcn_s_wait_tensorcnt)
#define USE_TDM 1
#else
#define USE_TDM 0
#endif

__device__ __forceinline__ unsigned long long wave_min_u64(unsigned long long v) {
#pragma unroll
  for (int m = 16; m > 0; m >>= 1) {
    unsigned long long o = __shfl_xor(v, m, 32);
    v = (o < v) ? o : v;
  }
  return v;
}

__global__ __launch_bounds__(THREADS, 1)
void prim_mst_gfx1250_kernel(const float* __restrict__ dist, int* __restrict__ out) {
  __shared__ float rowBuf[NPTS];                    // 16 KB row staging (TDM dest)
  __shared__ unsigned long long redWS[THREADS / 32];
  __shared__ int jslot;

  const int tid = (int)threadIdx.x;
  const int lane = tid & 31;
  const int wid = tid >> 5;

  // ---- init: mind = distances[0,:], parent = 0, node 0 in tree ----
  float mind[PER];
  int par[PER];
  unsigned usedMask = 0u;
  {
    const float4* d4 = (const float4*)dist;
    float4 r0 = d4[tid];                            // coalesced 16B per thread
    mind[0] = r0.x; mind[1] = r0.y; mind[2] = r0.z; mind[3] = r0.w;
    par[0] = par[1] = par[2] = par[3] = 0;
    if (tid == 0) usedMask = 1u;                    // node 0; mind[0] is +inf in input
  }

  for (int s = 0; s < NPTS - 1; ++s) {
    // ---- phase A: local + intra-wave argmin on packed (valbits<<32 | idx) ----
    unsigned long long best = ~0ull;
#pragma unroll
    for (int i = 0; i < PER; ++i) {
      unsigned long long k =
          ((unsigned long long)__float_as_uint(mind[i]) << 32) |
          (unsigned)(tid * PER + i);
      best = (k < best) ? k : best;
    }
    best = wave_min_u64(best);
    if (lane == 0) redWS[wid] = best;
    __syncthreads();

    // ---- phase B: wave 0 final reduce, publish j, DMA row j into LDS ----
    if (wid == 0) {
      unsigned long long b = redWS[lane];           // exactly 32 partials
      b = wave_min_u64(b);
      int j = (int)(unsigned)(b & 0xffffffffu);
      if (lane == 0) jslot = j;
#if USE_TDM
      int ju = __builtin_amdgcn_readfirstlane(j);
      unsigned long long gaddr =
          (unsigned long long)(uintptr_t)dist +
          (unsigned long long)ju * (unsigned long long)(NPTS * 4);
      unsigned lds_addr = (unsigned)(unsigned long long)(void*)rowBuf; // low 32b = LDS offset
      // D# group 0: count=1 | lds_addr | global_addr | type=2
      u32x4 g0;
      g0[0] = 1u;
      g0[1] = lds_addr;
      g0[2] = (unsigned)gaddr;
      g0[3] = ((unsigned)(gaddr >> 32) & 0x01ffffffu) | (2u << 30);
      // D# group 1: data_size=4B; tensor_dim0=4096, tensor_dim1=1;
      //             tile_dim0=4096, tile_dim1=1; tensor_dim0_stride=4096
      i32x8 g1;
      g1[0] = (int)0x00020000;   // data_size=2 (4 bytes), no mask/pad/iterate
      g1[1] = (int)0x10000000;   // tensor_dim0[15:0]=4096 in [31:16]
      g1[2] = (int)0x00010000;   // tensor_dim1=1 in [31:16]
      g1[3] = (int)0x10000000;   // tile_dim0=4096 in [31:16]
      g1[4] = 1;                 // tile_dim1=1, tile_dim2=0
      g1[5] = NPTS;              // tensor_dim0_stride low 32
      g1[6] = 0;
      g1[7] = 0;
      i32x4 gz = {0, 0, 0, 0};
#if defined(__clang_major__) && __clang_major__ >= 23
      i32x8 gz8 = {0, 0, 0, 0, 0, 0, 0, 0};
      __builtin_amdgcn_tensor_load_to_lds(g0, g1, gz, gz, gz8, 0);
#else
      __builtin_amdgcn_tensor_load_to_lds(g0, g1, gz, gz, 0);
#endif
      __builtin_amdgcn_s_wait_tensorcnt(0);
#endif
    }
    __syncthreads();

    const int j = jslot;
    // owning thread emits the edge (pre-update parent) and retires node j
    if ((j >> 2) == tid) {
      const int sl = j & 3;
      out[2 * s] = par[sl];
      out[2 * s + 1] = j;
      mind[sl] = __uint_as_float(0x7f800000u);      // +inf
      usedMask |= (1u << sl);
    }

#if USE_TDM
    const float4* row4 = (const float4*)rowBuf;     // ds_load_b128
    float4 dj4 = row4[tid];
#else
    const float4* row4 = (const float4*)(dist + (size_t)j * NPTS);
    float4 dj4 = row4[tid];
#endif
    float dj[PER] = {dj4.x, dj4.y, dj4.z, dj4.w};
#pragma unroll
    for (int i = 0; i < PER; ++i) {
      bool upd = (((usedMask >> i) & 1u) == 0u) && (dj[i] < mind[i]);
      mind[i] = upd ? dj[i] : mind[i];
      par[i] = upd ? j : par[i];
    }
    // no extra barrier needed: phase-A's barrier orders rowBuf/redWS reuse
  }
}

extern "C" void kernel_launch(void* const* d_in, const int* in_sizes, int n_in,
                              void* d_out, int out_size, void* d_ws, size_t ws_size,
                              hipStream_t stream) {
  (void)in_sizes; (void)n_in; (void)d_ws; (void)ws_size; (void)out_size;
  const float* dist = (const float*)d_in[0];
  int* out = (int*)d_out;
  prim_mst_gfx1250_kernel<<<1, THREADS, 0, stream>>>(dist, out);
}